// QMLP_63823214018945
// MI455X (gfx1250) — compile-verified
//
#include <hip/hip_runtime.h>

#define N_QUBITS   16
#define BATCH      256
#define STATE_N    65536                                   // 2^16 amplitudes
#define STATE_BYTES ((size_t)BATCH * STATE_N * sizeof(float2))   // 128 MB
#define GATES_FLOATS (2 * BATCH * N_QUBITS * 8)            // 2 layers * B * 16 * (4 cplx)
#define GATES_BYTES ((size_t)GATES_FLOATS * sizeof(float))

#define CHUNK_AMPS 8192          // 2^13: contiguous subspace of qubits 0..12
#define NCHUNK     (STATE_N / CHUNK_AMPS)

typedef __attribute__((ext_vector_type(2))) float v2f;
typedef __attribute__((ext_vector_type(8))) float v8f;

// ---------------------------------------------------------------------------
// K1: AvgPool2d(7) feature + fused U = Rot(phi,theta,omega) * RX(angle)
//     for both layers. One thread per (sample, qubit).
// ---------------------------------------------------------------------------
__global__ __launch_bounds__(256) void qmlp_gates(
    const float* __restrict__ x, const float* __restrict__ w0,
    const float* __restrict__ w1, float* __restrict__ gatesU)
{
  int g = blockIdx.x * 256 + threadIdx.x;   // 0..4095
  int b = g >> 4;
  int q = g & 15;

  // feature q = mean of 7x7 block (q>>2, q&3) of the 28x28 image
  int r0 = (q >> 2) * 7, c0 = (q & 3) * 7;
  const float* img = x + (size_t)b * 784;
  float acc = 0.f;
  for (int r = 0; r < 7; ++r)
    for (int c = 0; c < 7; ++c)
      acc += img[(r0 + r) * 28 + (c0 + c)];
  float ang = acc * (1.0f / 49.0f);

  float ca = cosf(0.5f * ang);     // RX diag
  float sa = sinf(0.5f * ang);     // RX offdiag is (-i*sa)

  for (int layer = 0; layer < 2; ++layer) {
    const float* w = (layer == 0) ? w0 : w1;
    float phi = w[q * 3 + 0], th = w[q * 3 + 1], om = w[q * 3 + 2];
    float ct = cosf(0.5f * th), st = sinf(0.5f * th);
    float p = 0.5f * (phi + om), m = 0.5f * (phi - om);
    float cp = cosf(p), sp = sinf(p), cm = cosf(m), sm = sinf(m);
    // Rot = [[a, b], [c, d]]
    float Ar =  ct * cp, Ai = -ct * sp;   // a = e^{-ip} ct
    float Br = -st * cm, Bi = -st * sm;   // b = -e^{+im} st
    float Cr =  st * cm, Ci = -st * sm;   // c = e^{-im} st
    float Dr =  ct * cp, Di =  ct * sp;   // d = e^{+ip} ct
    // F = Rot * RX ; (-i*sa)*(x+iy) = (sa*y, -sa*x)
    float F00r = Ar * ca + sa * Bi,  F00i = Ai * ca - sa * Br;
    float F01r = sa * Ai + Br * ca,  F01i = -sa * Ar + Bi * ca;
    float F10r = Cr * ca + sa * Di,  F10i = Ci * ca - sa * Dr;
    float F11r = sa * Ci + Dr * ca,  F11i = -sa * Cr + Di * ca;

    float* dst = gatesU + (((size_t)layer * BATCH + b) * N_QUBITS + q) * 8;
    dst[0] = F00r; dst[1] = F00i; dst[2] = F01r; dst[3] = F01i;
    dst[4] = F10r; dst[5] = F10i; dst[6] = F11r; dst[7] = F11i;
  }
}

// ---------------------------------------------------------------------------
// K2: initialize |0...0> for every sample
// ---------------------------------------------------------------------------
__global__ __launch_bounds__(256) void qmlp_init(float2* __restrict__ state)
{
  size_t i = (size_t)blockIdx.x * 256 + threadIdx.x;
  float2 v;
  v.x = ((i & (STATE_N - 1)) == 0) ? 1.0f : 0.0f;
  v.y = 0.0f;
  state[i] = v;
}

// ---------------------------------------------------------------------------
// Register-resident helpers for stage 2 (32-amp subcube per thread).
// Local bit 0 <-> qubit 0, local bits 1..4 <-> qubits 12..15.
// ---------------------------------------------------------------------------
__device__ __forceinline__ void gate1q_reg(float2* a, int lb, const float* U)
{
  float U00r = U[0], U00i = U[1], U01r = U[2], U01i = U[3];
  float U10r = U[4], U10i = U[5], U11r = U[6], U11i = U[7];
  int m = 1 << lb;
#pragma unroll
  for (int e = 0; e < 32; ++e) {
    if (e & m) continue;
    int e1 = e | m;
    float2 a0 = a[e], a1 = a[e1];
    float2 n0, n1;
    n0.x = U00r * a0.x - U00i * a0.y + U01r * a1.x - U01i * a1.y;
    n0.y = U00r * a0.y + U00i * a0.x + U01r * a1.y + U01i * a1.x;
    n1.x = U10r * a0.x - U10i * a0.y + U11r * a1.x - U11i * a1.y;
    n1.y = U10r * a0.y + U10i * a0.x + U11r * a1.y + U11i * a1.x;
    a[e] = n0; a[e1] = n1;
  }
}

__device__ __forceinline__ void crx_reg(float2* a, int cb, int tb, float c, float s)
{
  int cm = 1 << cb, tm = 1 << tb;
#pragma unroll
  for (int e = 0; e < 32; ++e) {
    if (!(e & cm) || (e & tm)) continue;   // control=1, target=0
    int e1 = e | tm;
    float2 a0 = a[e], a1 = a[e1];
    float2 n0, n1;
    // [ c  -i s ; -i s  c ] ; (-i s)*(x+iy) = (s*y, -s*x)
    n0.x = c * a0.x + s * a1.y;  n0.y = c * a0.y - s * a1.x;
    n1.x = c * a1.x + s * a0.y;  n1.y = c * a1.y - s * a0.x;
    a[e] = n0; a[e1] = n1;
  }
}

// ---------------------------------------------------------------------------
// K3: full circuit evolution, gate-fused. One workgroup (1024 threads) per
//     sample. Per layer:
//       Stage 1: qubits 0..12 are block-diagonal over contiguous 8192-amp
//                chunks -> load chunk to LDS (64KB) once, apply 1q(q=0..12)
//                and CRX(0..11) there, write back. 1 global pass for 25 gates.
//       Stage 2: qubits {0,12..15} are block-diagonal over 32-amp subcubes ->
//                each thread gathers a subcube into registers, applies
//                1q(q=13..15) + CRX(12..15) fully unrolled, scatters back.
//     Sequential CRX-chain order (0..15) is preserved; deferred 1q gates
//     commute with every earlier gate (disjoint wires).
// ---------------------------------------------------------------------------
__global__ __launch_bounds__(1024) void qmlp_evolve(
    float2* __restrict__ state, const float* __restrict__ gatesU,
    const float* __restrict__ x0, const float* __restrict__ x1)
{
  __shared__ float2 L[CHUNK_AMPS];          // 64 KB of the 320 KB/WGP LDS
  int b = blockIdx.x;
  int tid = threadIdx.x;
  float2* S = state + (size_t)b * STATE_N;

  for (int layer = 0; layer < 2; ++layer) {
    const float* gl = gatesU + ((size_t)layer * BATCH + b) * N_QUBITS * 8;
    const float* xw = (layer == 0) ? x0 : x1;

    // ---------------- Stage 1: qubits 0..12 in LDS ----------------
    for (int chunk = 0; chunk < NCHUNK; ++chunk) {
      const size_t base = (size_t)chunk * CHUNK_AMPS;
      // load chunk (b128 vectors): 4096 float4 = 8192 amps
      {
        const float4* src = (const float4*)(S + base);
        float4* dst = (float4*)L;
        for (int i = tid; i < CHUNK_AMPS / 2; i += 1024) dst[i] = src[i];
      }
      __syncthreads();

      // fused 1q gates on qubits 0..12
      for (int q = 0; q <= 12; ++q) {
        const float* U = gl + q * 8;
        float U00r = U[0], U00i = U[1], U01r = U[2], U01i = U[3];
        float U10r = U[4], U10i = U[5], U11r = U[6], U11i = U[7];
        int mask = 1 << q, low = mask - 1;
        for (int p = tid; p < (CHUNK_AMPS >> 1); p += 1024) {
          int i0 = ((p & ~low) << 1) | (p & low);
          int i1 = i0 | mask;
          float2 a0 = L[i0], a1 = L[i1];
          float2 n0, n1;
          n0.x = U00r * a0.x - U00i * a0.y + U01r * a1.x - U01i * a1.y;
          n0.y = U00r * a0.y + U00i * a0.x + U01r * a1.y + U01i * a1.x;
          n1.x = U10r * a0.x - U10i * a0.y + U11r * a1.x - U11i * a1.y;
          n1.y = U10r * a0.y + U10i * a0.x + U11r * a1.y + U11i * a1.x;
          L[i0] = n0; L[i1] = n1;
        }
        __syncthreads();
      }

      // CRX chain g = 0..11 (control g, target g+1), sequential
      for (int g = 0; g <= 11; ++g) {
        float th = xw[g];
        float c = cosf(0.5f * th), s = sinf(0.5f * th);
        int lom = (1 << g) - 1, him = (1 << (g + 1)) - 1;
        int cbit = 1 << g, tbit = 1 << (g + 1);
        for (int p = tid; p < (CHUNK_AMPS >> 2); p += 1024) {
          int t = ((p & ~lom) << 1) | (p & lom);   // insert 0 at bit g
          t = ((t & ~him) << 1) | (t & him);       // insert 0 at bit g+1
          int i0 = t | cbit;                       // control=1, target=0
          int i1 = i0 | tbit;
          float2 a0 = L[i0], a1 = L[i1];
          float2 n0, n1;
          n0.x = c * a0.x + s * a1.y;  n0.y = c * a0.y - s * a1.x;
          n1.x = c * a1.x + s * a0.y;  n1.y = c * a1.y - s * a0.x;
          L[i0] = n0; L[i1] = n1;
        }
        __syncthreads();
      }

      // write chunk back
      {
        float4* dst = (float4*)(S + base);
        const float4* src = (const float4*)L;
        for (int i = tid; i < CHUNK_AMPS / 2; i += 1024) dst[i] = src[i];
      }
      __syncthreads();
    }

    // ---------------- Stage 2: qubits {0,12,13,14,15} in registers ---------
    // subcube index j over bits 1..11 (2048 subcubes, 2 per thread);
    // element e: bit0 -> qubit 0, bits 1..4 -> qubits 12..15.
    float c12 = cosf(0.5f * xw[12]), s12 = sinf(0.5f * xw[12]);
    float c13 = cosf(0.5f * xw[13]), s13 = sinf(0.5f * xw[13]);
    float c14 = cosf(0.5f * xw[14]), s14 = sinf(0.5f * xw[14]);
    float c15 = cosf(0.5f * xw[15]), s15 = sinf(0.5f * xw[15]);
    for (int rep = 0; rep < 2; ++rep) {
      int j = rep * 1024 + tid;               // 0..2047
      float2 a[32];
#pragma unroll
      for (int e = 0; e < 32; ++e)
        a[e] = S[(size_t)((j << 1) | (e & 1) | ((e >> 1) << 12))];

      gate1q_reg(a, 2, gl + 13 * 8);          // 1q on qubit 13 (local bit 2)
      gate1q_reg(a, 3, gl + 14 * 8);          // qubit 14
      gate1q_reg(a, 4, gl + 15 * 8);          // qubit 15
      crx_reg(a, 1, 2, c12, s12);             // CRX(12,13)
      crx_reg(a, 2, 3, c13, s13);             // CRX(13,14)
      crx_reg(a, 3, 4, c14, s14);             // CRX(14,15)
      crx_reg(a, 4, 0, c15, s15);             // CRX(15,0)

#pragma unroll
      for (int e = 0; e < 32; ++e)
        S[(size_t)((j << 1) | (e & 1) | ((e >> 1) << 12))] = a[e];
    }
    __syncthreads();                          // make layer fully visible
  }
}

// ---------------------------------------------------------------------------
// K4: Pauli-Z expectations as a GEMM: z = probs(256x65536) @ sign(65536x16)
//     using V_WMMA_F32_16X16X4_F32, K accumulated in chunks of 4.
//     16 blocks (16 samples each) x 8 waves; waves split K, LDS-reduced.
//     A fragment (16x4 f32, 2 VGPRs): lanes 0-15 carry K=k0,k0+1;
//     lanes 16-31 carry K=k0+2,k0+3; M = lane&15 = sample-in-tile.
//     B fragment generated from bit patterns (no memory traffic).
// ---------------------------------------------------------------------------
__global__ __launch_bounds__(256) void qmlp_expect(
    const float2* __restrict__ state, float* __restrict__ z)
{
  __shared__ float red[8][32][8];
  int b0 = blockIdx.x * 16;
  int lane = threadIdx.x & 31;
  int wave = threadIdx.x >> 5;
  int sample = b0 + (lane & 15);
  int khalf = (lane >> 4) << 1;          // 0 or 2
  int n = lane & 15;                     // qubit column this lane covers in B
  const float2* sp = state + (size_t)sample * STATE_N;

  v8f acc = {};
  int kbeg = wave * (STATE_N / 8), kend = kbeg + (STATE_N / 8);
  for (int k0 = kbeg; k0 < kend; k0 += 4) {
    int ka = k0 + khalf;
    float2 s0 = sp[ka], s1 = sp[ka + 1];
    v2f A;
    A.x = s0.x * s0.x + s0.y * s0.y;     // prob(sample, ka)
    A.y = s1.x * s1.x + s1.y * s1.y;     // prob(sample, ka+1)
    v2f Bv;
    Bv.x = 1.0f - 2.0f * (float)((ka >> n) & 1);
    Bv.y = 1.0f - 2.0f * (float)(((ka + 1) >> n) & 1);
    acc = __builtin_amdgcn_wmma_f32_16x16x4_f32(
        /*neg_a=*/false, A, /*neg_b=*/false, Bv,
        /*c_mod=*/(short)0, acc, /*reuse_a=*/false, /*reuse_b=*/false);
  }

  for (int r = 0; r < 8; ++r) red[wave][lane][r] = acc[r];
  __syncthreads();
  if (wave == 0) {
    for (int r = 0; r < 8; ++r) {
      float sum = 0.f;
      for (int w = 0; w < 8; ++w) sum += red[w][lane][r];
      int M = r + ((lane >> 4) << 3);          // D row = sample-in-tile
      z[(size_t)(b0 + M) * N_QUBITS + (lane & 15)] = sum;
    }
  }
}

// ---------------------------------------------------------------------------
// K5: logits = z @ fcW^T + fcb, then log_softmax. One thread per sample.
// ---------------------------------------------------------------------------
__global__ __launch_bounds__(256) void qmlp_head(
    const float* __restrict__ z, const float* __restrict__ fcW,
    const float* __restrict__ fcb, float* __restrict__ out)
{
  int b = threadIdx.x;
  float logits[10];
  float mx = -1e30f;
  for (int j = 0; j < 10; ++j) {
    float s = fcb[j];
    for (int i = 0; i < 16; ++i) s += z[b * 16 + i] * fcW[j * 16 + i];
    logits[j] = s;
    mx = fmaxf(mx, s);
  }
  float se = 0.f;
  for (int j = 0; j < 10; ++j) se += expf(logits[j] - mx);
  float lse = logf(se);
  for (int j = 0; j < 10; ++j) out[b * 10 + j] = logits[j] - mx - lse;
}

// ---------------------------------------------------------------------------
extern "C" void kernel_launch(void* const* d_in, const int* in_sizes, int n_in,
                              void* d_out, int out_size, void* d_ws, size_t ws_size,
                              hipStream_t stream)
{
  const float* x   = (const float*)d_in[0];
  const float* w0  = (const float*)d_in[1];
  const float* w1  = (const float*)d_in[2];
  const float* x0  = (const float*)d_in[3];
  const float* x1  = (const float*)d_in[4];
  const float* fcW = (const float*)d_in[5];
  const float* fcb = (const float*)d_in[6];
  float* out = (float*)d_out;

  char* ws = (char*)d_ws;
  float2* state = (float2*)ws;                                 // 128 MB
  float* gatesU = (float*)(ws + STATE_BYTES);                  // 256 KB
  float* zbuf   = (float*)(ws + STATE_BYTES + GATES_BYTES);    // 16 KB

  qmlp_gates <<<16, 256, 0, stream>>>(x, w0, w1, gatesU);
  qmlp_init  <<<(BATCH * STATE_N) / 256, 256, 0, stream>>>(state);
  qmlp_evolve<<<BATCH, 1024, 0, stream>>>(state, gatesU, x0, x1);
  qmlp_expect<<<16, 256, 0, stream>>>(state, zbuf);
  qmlp_head  <<<1, 256, 0, stream>>>(zbuf, fcW, fcb, out);
}